// SAGE_DIFFPOOL_57604101374729
// MI455X (gfx1250) — compile-verified
//
#include <hip/hip_runtime.h>

typedef __attribute__((ext_vector_type(2))) float f32x2;
typedef __attribute__((ext_vector_type(8))) float f32x8;

static constexpr int Bb = 8, Nn = 2048, BN = Bb * Nn; // 16384 rows

// ---------------- workspace layout (float offsets) ----------------
static constexpr size_t OFF_DINVW = 0;
static constexpr size_t OFF_DINVB = OFF_DINVW + BN;
static constexpr size_t OFF_FILLW = OFF_DINVB + BN;
static constexpr size_t OFF_FILLB = OFF_FILLW + BN;
static constexpr size_t OFF_XCAT  = OFF_FILLB + BN;                    // BN*144
static constexpr size_t OFF_YBUF  = OFF_XCAT + (size_t)BN * 144;       // BN*144
static constexpr size_t OFF_SCAT  = OFF_YBUF + (size_t)BN * 144;       // BN*160 (aliased as pool partials later)
static constexpr size_t OFF_X13   = OFF_SCAT + (size_t)BN * 160;       // BN*30
static constexpr size_t OFF_PARTS = OFF_X13 + (size_t)BN * 30;         // 64*288
static constexpr size_t OFF_ST1   = OFF_PARTS + 64 * 288;              // 128
static constexpr size_t OFF_ST2   = OFF_ST1 + 128;                     // 128
static constexpr size_t OFF_ST3   = OFF_ST2 + 128;                     // 288
static constexpr size_t OFF_P1X   = OFF_ST3 + 288;                     // 8*100*30
static constexpr size_t OFF_P1ADJ = OFF_P1X + 24000;                   // 8*100*100
static constexpr size_t OFF_A2N   = OFF_P1ADJ + 80000;                 // 80000
static constexpr size_t OFF_X2A   = OFF_A2N + 80000;                   // 24000
static constexpr size_t OFF_Z2    = OFF_X2A + 24000;                   // 24000
static constexpr size_t OFF_X2B   = OFF_Z2 + 24000;                    // 24000
static constexpr size_t OFF_S2P   = OFF_X2B + 24000;                   // 3*480
static constexpr size_t OFF_X1MAX = OFF_S2P + 1440;                    // 8*96 (uint)
static constexpr size_t OFF_X2MAX = OFF_X1MAX + 768;                   // 8*96 (uint)
static constexpr size_t OFF_P1PART = OFF_SCAT;                         // 128*13000 aliases SCAT (dead by then)

// monotone float<->uint encoding for atomic max
static __device__ __forceinline__ unsigned encf(float f) {
  unsigned u = __float_as_uint(f);
  return (u & 0x80000000u) ? ~u : (u | 0x80000000u);
}
static __device__ __forceinline__ float decf(unsigned k) {
  unsigned u = (k & 0x80000000u) ? (k & 0x7fffffffu) : ~k;
  return __uint_as_float(u);
}

// ---------------- init: only the atomic-max buffers need zeroing ----------------
__global__ void k_init(unsigned* x1max, unsigned* x2max) {
  int t = blockIdx.x * 256 + threadIdx.x;
  if (t < 768) { x1max[t] = 0u; x2max[t] = 0u; }
}

// ---------------- per-row degrees / fills for weighted and binary adjacency ----------------
__global__ void __launch_bounds__(256)
k_deg(const float* __restrict__ adj, float* dinvW, float* dinvB, float* fillW, float* fillB) {
  int rowg = blockIdx.x;               // 0..BN-1
  int n = rowg & (Nn - 1);
  const float* ar = adj + (size_t)rowg * Nn;
  float sw = 0.f, sb = 0.f;
  for (int j = threadIdx.x; j < Nn; j += 256) {
    float a = ar[j];
    sw += a;
    sb += (a > 0.f) ? 1.f : 0.f;
  }
  __shared__ float rw[256], rb[256];
  rw[threadIdx.x] = sw; rb[threadIdx.x] = sb;
  __syncthreads();
  for (int s = 128; s > 0; s >>= 1) {
    if (threadIdx.x < s) { rw[threadIdx.x] += rw[threadIdx.x + s]; rb[threadIdx.x] += rb[threadIdx.x + s]; }
    __syncthreads();
  }
  if (threadIdx.x == 0) {
    float d  = ar[n];
    float fw = (d == 0.f) ? 1.f : 0.f;
    float fb = (d > 0.f) ? 0.f : 1.f;
    float degw = rw[0] + fw, degb = rb[0] + fb;
    dinvW[rowg] = degw > 0.f ? rsqrtf(degw) : 0.f;
    dinvB[rowg] = degb > 0.f ? rsqrtf(degb) : 0.f;
    fillW[rowg] = fw; fillB[rowg] = fb;
  }
}

// ---------------- initial 3->30 projection, pre-scaled by dinv, packed [w32|b32] ----------------
__global__ void __launch_bounds__(256)
k_feat_proj(const float* __restrict__ x, const float* __restrict__ Win,
            const float* __restrict__ dinvW, const float* __restrict__ dinvB,
            float* __restrict__ XC) {
  int rowg = blockIdx.x * 256 + threadIdx.x;
  float x0 = x[rowg * 3 + 0], x1 = x[rowg * 3 + 1], x2 = x[rowg * 3 + 2];
  float dw = dinvW[rowg], db = dinvB[rowg];
  float* o = XC + (size_t)rowg * 64;
  for (int c = 0; c < 30; c++) {
    float vw = x0 * Win[c] + x1 * Win[30 + c] + x2 * Win[60 + c];
    float vb = x0 * Win[90 + c] + x1 * Win[120 + c] + x2 * Win[150 + c];
    o[c] = dw * vw;
    o[32 + c] = db * vb;
  }
  o[30] = 0.f; o[31] = 0.f; o[62] = 0.f; o[63] = 0.f;
}

// ---------------- the big batched GEMM: Y = Araw @ X (cols<CW raw adj value, else binarized) ----
template <int C, int CW>
__global__ void __launch_bounds__(256)
k_gemm(const float* __restrict__ adj, const float* __restrict__ X, float* __restrict__ Y) {
  constexpr int KT = (C <= 64) ? 128 : 64;     // LDS chunk of K rows
  constexpr int NT = C / 16;                   // WMMA column tiles
  static_assert((KT * C) % 1024 == 0, "coop load");
  __shared__ float lds[KT * C];
  const int b    = blockIdx.x;                 // batch
  const int rt   = blockIdx.y;                 // 128-row tile
  const int wave = threadIdx.x >> 5;
  const int lane = threadIdx.x & 31;
  const int hi   = lane >> 4;                  // lane half selects K-pair
  const int l16  = lane & 15;
  const int rowA = rt * 128 + wave * 16 + l16; // A-fragment row for this lane
  const float* __restrict__ adjRow = adj + ((size_t)b * Nn + rowA) * Nn;
  const float* __restrict__ Xb     = X + (size_t)b * Nn * C;

  f32x8 acc[NT];
#pragma unroll
  for (int t = 0; t < NT; ++t) acc[t] = f32x8{0.f, 0.f, 0.f, 0.f, 0.f, 0.f, 0.f, 0.f};

  for (int j0 = 0; j0 < Nn; j0 += KT) {
    __syncthreads();
    const float4* __restrict__ src = (const float4*)(Xb + (size_t)j0 * C);
    float4* dst = (float4*)lds;
#pragma unroll
    for (int i = 0; i < (KT * C) / 1024; ++i)
      dst[threadIdx.x + i * 256] = src[threadIdx.x + i * 256];
    __syncthreads();
    for (int jj = 0; jj < KT; jj += 4) {
      const int kb = jj + 2 * hi;
      // A frag (f32 16x4): VGPR0=K(kb), VGPR1=K(kb+1) -> one b64 per lane
      f32x2 aW = *(const f32x2*)(adjRow + j0 + kb);
      f32x2 aB;
      aB.x = aW.x > 0.f ? 1.f : 0.f;
      aB.y = aW.y > 0.f ? 1.f : 0.f;
#pragma unroll
      for (int t = 0; t < NT; ++t) {
        f32x2 bf;                                 // B frag: VGPRv = K(kb+v), N = l16
        bf.x = lds[(kb + 0) * C + t * 16 + l16];
        bf.y = lds[(kb + 1) * C + t * 16 + l16];
        f32x2 af = (t * 16 < CW) ? aW : aB;
        acc[t] = __builtin_amdgcn_wmma_f32_16x16x4_f32(
            false, af, false, bf, (short)0, acc[t], false, false);
      }
    }
  }
#pragma unroll
  for (int t = 0; t < NT; ++t)
#pragma unroll
    for (int r = 0; r < 8; ++r) {
      int orow = rt * 128 + wave * 16 + hi * 8 + r;   // C/D layout: vgpr r -> M=r / M=r+8
      Y[((size_t)b * Nn + orow) * C + t * 16 + l16] = acc[t][r];
    }
}

// ---------------- epilogue: z = dinv*(Y + fill*X') + bias, per-block stat partials ----------------
template <int C, int CW>
__global__ void __launch_bounds__(256)
k_epi(float* __restrict__ Y, const float* __restrict__ XC,
      const float* __restrict__ dinvW, const float* __restrict__ dinvB,
      const float* __restrict__ fillW, const float* __restrict__ fillB,
      const float* __restrict__ biasW, int nW, const float* __restrict__ biasB, int nB,
      float* __restrict__ parts) {
  const int c = threadIdx.x;
  const int r0 = blockIdx.x * 256;
  if (c >= C) return;
  const bool isW = (c < CW);
  const int cl = isW ? c : c - CW;
  float bias = 0.f;
  if (isW) { if (cl < nW) bias = biasW[cl]; }
  else     { if (cl < nB) bias = biasB[cl]; }
  const float* di = isW ? dinvW : dinvB;
  const float* fi = isW ? fillW : fillB;
  float sum = 0.f, sq = 0.f;
  for (int r = 0; r < 256; r++) {
    int rowg = r0 + r;
    size_t idx = (size_t)rowg * C + c;
    float z = di[rowg] * (Y[idx] + fi[rowg] * XC[idx]) + bias;
    Y[idx] = z;
    sum += z; sq += z * z;
  }
  parts[blockIdx.x * (2 * C) + c] = sum;
  parts[blockIdx.x * (2 * C) + C + c] = sq;
}

__global__ void k_stats_reduce(const float* __restrict__ parts, float* __restrict__ stats, int C) {
  int c = threadIdx.x;
  if (c >= C) return;
  float s = 0.f, q = 0.f;
  for (int b = 0; b < 64; b++) { s += parts[b * 2 * C + c]; q += parts[b * 2 * C + C + c]; }
  float m = s * (1.f / 16384.f);
  float v = q * (1.f / 16384.f) - m * m;
  stats[c] = m;
  stats[C + c] = rsqrtf(v + 1e-5f);
}

// ---------------- BN + node-max + projection to next layer's scaled input ----------------
__global__ void __launch_bounds__(256)
k_bn1(const float* __restrict__ Z, const float* __restrict__ st,
      const float* __restrict__ g30, const float* __restrict__ be30,
      const float* __restrict__ W3030, const float* __restrict__ dinvW, const float* __restrict__ dinvB,
      float* __restrict__ XC, float* __restrict__ SCAT, unsigned* __restrict__ x1max) {
  int rowg = blockIdx.x * 256 + threadIdx.x;
  int b = rowg >> 11;
  const float* zr = Z + (size_t)rowg * 64;
  float xv[30], sv[30];
  for (int c = 0; c < 30; c++) {
    xv[c] = (zr[c] - st[c]) * st[64 + c] * g30[c] + be30[c];
    sv[c] = (zr[32 + c] - st[32 + c]) * st[96 + c] * g30[90 + c] + be30[90 + c];
  }
  for (int c = 0; c < 30; c++) atomicMax(&x1max[b * 96 + c], encf(xv[c]));
  for (int c = 0; c < 30; c++) SCAT[(size_t)rowg * 160 + c] = sv[c];
  float dw = dinvW[rowg], db = dinvB[rowg];
  float* o = XC + (size_t)rowg * 64;
  for (int c = 0; c < 30; c++) {
    float aw = 0.f, ab = 0.f;
    for (int i = 0; i < 30; i++) { aw += xv[i] * W3030[i * 30 + c]; ab += sv[i] * W3030[1800 + i * 30 + c]; }
    o[c] = dw * aw; o[32 + c] = db * ab;
  }
  o[30] = 0.f; o[31] = 0.f; o[62] = 0.f; o[63] = 0.f;
}

__global__ void __launch_bounds__(256)
k_bn2(const float* __restrict__ Z, const float* __restrict__ st,
      const float* __restrict__ g30, const float* __restrict__ be30,
      const float* __restrict__ W3030, const float* __restrict__ Wp13,
      const float* __restrict__ dinvW, const float* __restrict__ dinvB,
      float* __restrict__ XC, float* __restrict__ SCAT, unsigned* __restrict__ x1max) {
  int rowg = blockIdx.x * 256 + threadIdx.x;
  int b = rowg >> 11;
  const float* zr = Z + (size_t)rowg * 64;
  float xv[30], sv[30];
  for (int c = 0; c < 30; c++) {
    xv[c] = (zr[c] - st[c]) * st[64 + c] * g30[30 + c] + be30[30 + c];
    sv[c] = (zr[32 + c] - st[32 + c]) * st[96 + c] * g30[120 + c] + be30[120 + c];
  }
  for (int c = 0; c < 30; c++) atomicMax(&x1max[b * 96 + 30 + c], encf(xv[c]));
  for (int c = 0; c < 30; c++) SCAT[(size_t)rowg * 160 + 30 + c] = sv[c];
  float dw = dinvW[rowg], db = dinvB[rowg];
  float* o = XC + (size_t)rowg * 144;
  for (int c = 0; c < 30; c++) {
    float aw = 0.f;
    for (int i = 0; i < 30; i++) aw += xv[i] * W3030[900 + i * 30 + c];
    o[c] = dw * aw;
  }
  o[30] = 0.f; o[31] = 0.f;
  for (int c = 0; c < 100; c++) {
    float ab = 0.f;
    for (int i = 0; i < 30; i++) ab += sv[i] * Wp13[i * 100 + c];
    o[32 + c] = db * ab;
  }
  for (int c = 132; c < 144; c++) o[c] = 0.f;
}

__global__ void __launch_bounds__(256)
k_bn3(const float* __restrict__ Z, const float* __restrict__ st,
      const float* __restrict__ g30, const float* __restrict__ be30,
      const float* __restrict__ g100, const float* __restrict__ be100,
      const float* __restrict__ Wfc, const float* __restrict__ bfc,
      float* __restrict__ X13, float* __restrict__ SCAT, float* __restrict__ XC,
      unsigned* __restrict__ x1max) {
  int rowg = blockIdx.x * 256 + threadIdx.x;
  int b = rowg >> 11;
  const float* zr = Z + (size_t)rowg * 144;
  float xv[30];
  for (int c = 0; c < 30; c++)
    xv[c] = (zr[c] - st[c]) * st[144 + c] * g30[60 + c] + be30[60 + c];
  for (int c = 0; c < 30; c++) {
    atomicMax(&x1max[b * 96 + 60 + c], encf(xv[c]));
    X13[(size_t)rowg * 30 + c] = xv[c];
  }
  float sv[100];
  for (int c = 0; c < 100; c++)
    sv[c] = (zr[32 + c] - st[32 + c]) * st[176 + c] * g100[c] + be100[c];
  for (int c = 0; c < 100; c++) SCAT[(size_t)rowg * 160 + 60 + c] = sv[c];
  // s1 = concat(s11,s12,s13) @ Wfc + bfc, then softmax
  const float* sc = SCAT + (size_t)rowg * 160;
  float s1[100];
  for (int c = 0; c < 100; c++) {
    float a = bfc[c];
    for (int i = 0; i < 60; i++) a += sc[i] * Wfc[i * 100 + c];
    for (int i = 0; i < 100; i++) a += sv[i] * Wfc[(60 + i) * 100 + c];
    s1[c] = a;
  }
  float mx = s1[0];
  for (int c = 1; c < 100; c++) mx = fmaxf(mx, s1[c]);
  float ssum = 0.f;
  for (int c = 0; c < 100; c++) { float e = __expf(s1[c] - mx); s1[c] = e; ssum += e; }
  float inv = 1.f / ssum;
  float* o = XC + (size_t)rowg * 112;
  for (int c = 0; c < 100; c++) o[c] = s1[c] * inv;
  for (int c = 100; c < 112; c++) o[c] = 0.f;
}

// ---------------- pooled reductions: p1_x = s^T x13, p1_adj = s^T (adj@s) ----------------
__global__ void __launch_bounds__(256)
k_pool_part(const float* __restrict__ S, const float* __restrict__ As,
            const float* __restrict__ X13, float* __restrict__ part) {
  __shared__ float sS[128][100];
  __shared__ float sR[128][132];   // [x13(30) | As(100)] padded
  const int chunk = blockIdx.x, b = blockIdx.y;
  const size_t base = (size_t)b * Nn + chunk * 128;
  for (int i = threadIdx.x; i < 128 * 100; i += 256) { int n = i / 100, c = i % 100; sS[n][c] = S[(base + n) * 112 + c]; }
  for (int i = threadIdx.x; i < 128 * 30;  i += 256) { int n = i / 30,  c = i % 30;  sR[n][c] = X13[(base + n) * 30 + c]; }
  for (int i = threadIdx.x; i < 128 * 100; i += 256) { int n = i / 100, c = i % 100; sR[n][30 + c] = As[(base + n) * 112 + c]; }
  __syncthreads();
  const int t = threadIdx.x;
  if (t < 200) {
    const int k = t >> 1, c0 = (t & 1) * 65;
    float acc[65];
#pragma unroll
    for (int i = 0; i < 65; i++) acc[i] = 0.f;
    for (int n = 0; n < 128; n++) {
      float sval = sS[n][k];
#pragma unroll
      for (int i = 0; i < 65; i++) acc[i] += sval * sR[n][c0 + i];
    }
    float* p = part + (size_t)(b * 16 + chunk) * 13000 + k * 130 + c0;
#pragma unroll
    for (int i = 0; i < 65; i++) p[i] = acc[i];
  }
}

__global__ void k_pool_reduce(const float* __restrict__ part, float* __restrict__ P1X, float* __restrict__ P1ADJ) {
  int b = blockIdx.y;
  int o = blockIdx.x * 256 + threadIdx.x;
  if (o >= 13000) return;
  float s = 0.f;
  for (int ch = 0; ch < 16; ch++) s += part[(size_t)(b * 16 + ch) * 13000 + o];
  int k = o / 130, c = o % 130;
  if (c < 30) P1X[(b * 100 + k) * 30 + c] = s;
  else        P1ADJ[(b * 100 + k) * 100 + (c - 30)] = s;
}

// ---------------- stage 2 (pooled 100-node graphs) ----------------
__global__ void __launch_bounds__(128)
k_s2norm(const float* __restrict__ P1ADJ, float* __restrict__ A2N) {
  __shared__ float dinv[100], fill[100];
  int b = blockIdx.x, k = threadIdx.x;
  if (k < 100) {
    const float* r = P1ADJ + (b * 100 + k) * 100;
    float s = 0.f;
    for (int l = 0; l < 100; l++) s += r[l];
    float d = r[k];
    float f = (d == 0.f) ? 1.f : 0.f;
    float deg = s + f;
    dinv[k] = deg > 0.f ? rsqrtf(deg) : 0.f;
    fill[k] = f;
  }
  __syncthreads();
  if (k < 100) {
    const float* r = P1ADJ + (b * 100 + k) * 100;
    float dk = dinv[k], fk = fill[k];
    for (int l = 0; l < 100; l++) {
      float v = r[l] + (l == k ? fk : 0.f);
      A2N[(b * 100 + k) * 100 + l] = dk * v * dinv[l];
    }
  }
}

__global__ void __launch_bounds__(128)
k_s2gcn(const float* __restrict__ Xin, const float* __restrict__ W, const float* __restrict__ bias,
        const float* __restrict__ A2N, float* __restrict__ Z2, float* __restrict__ part) {
  __shared__ float T[100][32];
  int b = blockIdx.x, k = threadIdx.x;
  if (k < 100) {
    const float* xr = Xin + (b * 100 + k) * 30;
    for (int c = 0; c < 30; c++) {
      float a = 0.f;
      for (int i = 0; i < 30; i++) a += xr[i] * W[i * 30 + c];
      T[k][c] = a;
    }
  }
  __syncthreads();
  float z[30];
  if (k < 100) {
    const float* ar = A2N + (b * 100 + k) * 100;
    for (int c = 0; c < 30; c++) z[c] = bias[c];
    for (int l = 0; l < 100; l++) {
      float a = ar[l];
      for (int c = 0; c < 30; c++) z[c] += a * T[l][c];
    }
    for (int c = 0; c < 30; c++) Z2[(b * 100 + k) * 30 + c] = z[c];
  }
  __syncthreads();
  if (k < 100) for (int c = 0; c < 30; c++) T[k][c] = z[c];
  __syncthreads();
  if (k < 30) {
    float s = 0.f, q = 0.f;
    for (int l = 0; l < 100; l++) { float v = T[l][k]; s += v; q += v * v; }
    part[b * 60 + k] = s;
    part[b * 60 + 30 + k] = q;
  }
}

__global__ void __launch_bounds__(128)
k_s2bn(const float* __restrict__ Z2, const float* __restrict__ part,
       const float* __restrict__ g, const float* __restrict__ be,
       float* __restrict__ Xout, unsigned* __restrict__ x2max, int colBase) {
  __shared__ float M[30], R[30];
  int b = blockIdx.x, k = threadIdx.x;
  if (k < 30) {
    float s = 0.f, q = 0.f;
    for (int bb = 0; bb < 8; bb++) { s += part[bb * 60 + k]; q += part[bb * 60 + 30 + k]; }
    float m = s * (1.f / 800.f);
    float v = q * (1.f / 800.f) - m * m;
    M[k] = m; R[k] = rsqrtf(v + 1e-5f);
  }
  __syncthreads();
  if (k < 100) {
    for (int c = 0; c < 30; c++) {
      float v = (Z2[(b * 100 + k) * 30 + c] - M[c]) * R[c] * g[c] + be[c];
      Xout[(b * 100 + k) * 30 + c] = v;
      atomicMax(&x2max[b * 96 + colBase + c], encf(v));
    }
  }
}

// ---------------- head MLP ----------------
__global__ void __launch_bounds__(256)
k_head(const unsigned* __restrict__ x1max, const unsigned* __restrict__ x2max,
       const float* __restrict__ W1, const float* __restrict__ b1,
       const float* __restrict__ W2, const float* __restrict__ b2, float* __restrict__ out) {
  __shared__ float feat[8][180];
  __shared__ float h[8][50];
  for (int i = threadIdx.x; i < 8 * 180; i += 256) {
    int b = i / 180, c = i % 180;
    feat[b][c] = (c < 90) ? decf(x1max[b * 96 + c]) : decf(x2max[b * 96 + c - 90]);
  }
  __syncthreads();
  for (int o = threadIdx.x; o < 400; o += 256) {
    int b = o / 50, j = o % 50;
    float a = b1[j];
    for (int i = 0; i < 180; i++) a += feat[b][i] * W1[i * 50 + j];
    h[b][j] = fmaxf(a, 0.f);
  }
  __syncthreads();
  if (threadIdx.x < 48) {
    int b = threadIdx.x / 6, j = threadIdx.x % 6;
    float a = b2[j];
    for (int i = 0; i < 50; i++) a += h[b][i] * W2[i * 6 + j];
    out[b * 6 + j] = a;
  }
}

// ---------------- launch ----------------
extern "C" void kernel_launch(void* const* d_in, const int* in_sizes, int n_in,
                              void* d_out, int out_size, void* d_ws, size_t ws_size,
                              hipStream_t stream) {
  (void)in_sizes; (void)n_in; (void)out_size; (void)ws_size;
  const float* x     = (const float*)d_in[0];
  const float* adj   = (const float*)d_in[1];
  const float* W_in  = (const float*)d_in[2];
  const float* W3030 = (const float*)d_in[3];
  const float* Wp13  = (const float*)d_in[4];
  const float* b30   = (const float*)d_in[5];
  const float* b100  = (const float*)d_in[6];
  const float* Wfc   = (const float*)d_in[7];
  const float* bfc   = (const float*)d_in[8];
  const float* W1    = (const float*)d_in[9];
  const float* b1    = (const float*)d_in[10];
  const float* W2    = (const float*)d_in[11];
  const float* b2    = (const float*)d_in[12];
  const float* g30   = (const float*)d_in[13];
  const float* be30  = (const float*)d_in[14];
  const float* g100  = (const float*)d_in[15];
  const float* be100 = (const float*)d_in[16];
  float* out = (float*)d_out;
  float* ws = (float*)d_ws;

  float* dinvW = ws + OFF_DINVW; float* dinvB = ws + OFF_DINVB;
  float* fillW = ws + OFF_FILLW; float* fillB = ws + OFF_FILLB;
  float* XCAT = ws + OFF_XCAT;   float* YBUF = ws + OFF_YBUF;
  float* SCAT = ws + OFF_SCAT;   float* X13 = ws + OFF_X13;
  float* PARTS = ws + OFF_PARTS;
  float* ST1 = ws + OFF_ST1; float* ST2 = ws + OFF_ST2; float* ST3 = ws + OFF_ST3;
  float* P1X = ws + OFF_P1X; float* P1ADJ = ws + OFF_P1ADJ; float* A2N = ws + OFF_A2N;
  float* X2A = ws + OFF_X2A; float* Z2 = ws + OFF_Z2; float* X2B = ws + OFF_X2B;
  float* S2P = ws + OFF_S2P;
  unsigned* X1MAX = (unsigned*)(ws + OFF_X1MAX);
  unsigned* X2MAX = (unsigned*)(ws + OFF_X2MAX);
  float* P1PART = ws + OFF_P1PART;

  k_init<<<3, 256, 0, stream>>>(X1MAX, X2MAX);
  k_deg<<<BN, 256, 0, stream>>>(adj, dinvW, dinvB, fillW, fillB);
  k_feat_proj<<<64, 256, 0, stream>>>(x, W_in, dinvW, dinvB, XCAT);

  // layer 1 (x11 | s11)
  k_gemm<64, 32><<<dim3(Bb, 16), 256, 0, stream>>>(adj, XCAT, YBUF);
  k_epi<64, 32><<<64, 256, 0, stream>>>(YBUF, XCAT, dinvW, dinvB, fillW, fillB,
                                        b30 + 0, 30, b30 + 90, 30, PARTS);
  k_stats_reduce<<<1, 256, 0, stream>>>(PARTS, ST1, 64);
  k_bn1<<<64, 256, 0, stream>>>(YBUF, ST1, g30, be30, W3030, dinvW, dinvB, XCAT, SCAT, X1MAX);

  // layer 2 (x12 | s12)
  k_gemm<64, 32><<<dim3(Bb, 16), 256, 0, stream>>>(adj, XCAT, YBUF);
  k_epi<64, 32><<<64, 256, 0, stream>>>(YBUF, XCAT, dinvW, dinvB, fillW, fillB,
                                        b30 + 30, 30, b30 + 120, 30, PARTS);
  k_stats_reduce<<<1, 256, 0, stream>>>(PARTS, ST2, 64);
  k_bn2<<<64, 256, 0, stream>>>(YBUF, ST2, g30, be30, W3030, Wp13, dinvW, dinvB, XCAT, SCAT, X1MAX);

  // layer 3 (x13 | s13), then fc_pool + softmax
  k_gemm<144, 32><<<dim3(Bb, 16), 256, 0, stream>>>(adj, XCAT, YBUF);
  k_epi<144, 32><<<64, 256, 0, stream>>>(YBUF, XCAT, dinvW, dinvB, fillW, fillB,
                                         b30 + 60, 30, b100, 100, PARTS);
  k_stats_reduce<<<1, 256, 0, stream>>>(PARTS, ST3, 144);
  k_bn3<<<64, 256, 0, stream>>>(YBUF, ST3, g30, be30, g100, be100, Wfc, bfc,
                                X13, SCAT, XCAT, X1MAX);

  // adj @ s (raw adjacency, 100 cols padded to 112)
  k_gemm<112, 112><<<dim3(Bb, 16), 256, 0, stream>>>(adj, XCAT, YBUF);

  // diffpool reductions
  k_pool_part<<<dim3(16, Bb), 256, 0, stream>>>(XCAT, YBUF, X13, P1PART);
  k_pool_reduce<<<dim3(51, Bb), 256, 0, stream>>>(P1PART, P1X, P1ADJ);

  // stage 2
  k_s2norm<<<Bb, 128, 0, stream>>>(P1ADJ, A2N);
  k_s2gcn<<<Bb, 128, 0, stream>>>(P1X, W3030 + 3 * 900, b30 + 150, A2N, Z2, S2P + 0);
  k_s2bn<<<Bb, 128, 0, stream>>>(Z2, S2P + 0, g30 + 150, be30 + 150, X2A, X2MAX, 0);
  k_s2gcn<<<Bb, 128, 0, stream>>>(X2A, W3030 + 4 * 900, b30 + 180, A2N, Z2, S2P + 480);
  k_s2bn<<<Bb, 128, 0, stream>>>(Z2, S2P + 480, g30 + 180, be30 + 180, X2B, X2MAX, 30);
  k_s2gcn<<<Bb, 128, 0, stream>>>(X2B, W3030 + 5 * 900, b30 + 210, A2N, Z2, S2P + 960);
  k_s2bn<<<Bb, 128, 0, stream>>>(Z2, S2P + 960, g30 + 210, be30 + 210, X2A, X2MAX, 60);

  k_head<<<1, 256, 0, stream>>>(X1MAX, X2MAX, W1, b1, W2, b2, out);
}